// BiMambaPlusBlock_26723286516242
// MI455X (gfx1250) — compile-verified
//
#include <hip/hip_runtime.h>
#include <hip/hip_bf16.h>

#define D_MODEL 1024
#define D_STATE 16
#define D_CONV  4
#define D_INNER 2048
#define DT_RANK 64
#define BATCH   2
#define SEQ     2048

typedef __attribute__((ext_vector_type(2))) float v2f;
typedef __attribute__((ext_vector_type(8))) float v8f;

// ---------------------------------------------------------------------------
// LayerNorm: one block per (b,l) row of 1024 features
// ---------------------------------------------------------------------------
__global__ __launch_bounds__(256) void layernorm_kernel(
    const float* __restrict__ x, const float* __restrict__ gamma,
    const float* __restrict__ beta, float* __restrict__ xn)
{
    __shared__ float s1[256];
    __shared__ float s2[256];
    const int row = blockIdx.x;
    const float* xr = x + (size_t)row * D_MODEL;
    float sum = 0.f, sumsq = 0.f;
    for (int i = threadIdx.x; i < D_MODEL; i += 256) {
        float v = xr[i];
        sum += v; sumsq += v * v;
    }
    s1[threadIdx.x] = sum; s2[threadIdx.x] = sumsq;
    __syncthreads();
    for (int st = 128; st > 0; st >>= 1) {
        if (threadIdx.x < st) {
            s1[threadIdx.x] += s1[threadIdx.x + st];
            s2[threadIdx.x] += s2[threadIdx.x + st];
        }
        __syncthreads();
    }
    const float mu   = s1[0] * (1.f / D_MODEL);
    const float var  = s2[0] * (1.f / D_MODEL) - mu * mu;
    const float rstd = rsqrtf(var + 1e-5f);
    float* xo = xn + (size_t)row * D_MODEL;
    for (int i = threadIdx.x; i < D_MODEL; i += 256)
        xo[i] = (xr[i] - mu) * rstd * gamma[i] + beta[i];
}

// ---------------------------------------------------------------------------
// Generic fp32 WMMA GEMM:  C[M,N] (+)= op( A[M,K] (opt. row-reversed per batch)
//                                          @ W[K,N] )
// One wave per block computes a (16*MT) x (16*NT) tile; MT*NT accumulators.
// A fragment reused across NT WMMAs, B fragment across MT WMMAs.
// Grid must satisfy M % (16*MT) == 0 and N % (16*NT) == 0 (compile-time tiling
// keeps the K-loop completely branch-free so EXEC stays all-ones for WMMA).
// epi: 0 = store, 1 = accumulate into C, 2 = softplus(acc + bias[n])
// revL: 0 = normal; else A rows are read as  b*revL + (revL-1-l)
// ---------------------------------------------------------------------------
template <int MT, int NT>
__global__ __launch_bounds__(32) void gemm_wmma_f32(
    const float* __restrict__ A, int lda,
    const float* __restrict__ W, int ldb,
    float* __restrict__ C, int ldc,
    const float* __restrict__ bias,
    int K, int revL, int epi)
{
    const int lane  = threadIdx.x & 31;
    const int half  = lane >> 4;     // 0: lanes 0-15, 1: lanes 16-31
    const int l16   = lane & 15;
    const int mBase = blockIdx.y * (16 * MT);
    const int nBase = blockIdx.x * (16 * NT);

    // A row pointers (A layout 16x4: both lane-halves cover M=0..15)
    const float* __restrict__ Arow[MT];
    #pragma unroll
    for (int mi = 0; mi < MT; ++mi) {
        int arow = mBase + 16 * mi + l16;
        if (revL > 0) {
            int b = arow / revL;
            int l = arow - b * revL;
            arow = b * revL + (revL - 1 - l);
        }
        Arow[mi] = A + (size_t)arow * lda;
    }

    v8f acc[MT][NT];
    #pragma unroll
    for (int mi = 0; mi < MT; ++mi)
        #pragma unroll
        for (int ni = 0; ni < NT; ++ni)
            acc[mi][ni] = (v8f){};

    // B fragment pointer: row (2*half) of current K quad, this lane's column.
    const float* __restrict__ Wp =
        W + (size_t)(2 * half) * ldb + nBase + l16;
    const size_t wStep = (size_t)4 * ldb;

    for (int k = 0; k < K; k += 4) {
        const int ka = k + 2 * half;                  // this lane-half's K pair
        v2f a[MT];
        #pragma unroll
        for (int mi = 0; mi < MT; ++mi) {
            a[mi].x = Arow[mi][ka];
            a[mi].y = Arow[mi][ka + 1];
        }
        if (k + 128 < K) {
            #pragma unroll
            for (int mi = 0; mi < MT; ++mi)
                __builtin_prefetch(Arow[mi] + k + 128, 0, 1);
        }
        v2f bv[NT];
        #pragma unroll
        for (int ni = 0; ni < NT; ++ni) {
            bv[ni].x = Wp[16 * ni];
            bv[ni].y = Wp[(size_t)ldb + 16 * ni];
        }
        #pragma unroll
        for (int mi = 0; mi < MT; ++mi)
            #pragma unroll
            for (int ni = 0; ni < NT; ++ni)
                acc[mi][ni] = __builtin_amdgcn_wmma_f32_16x16x4_f32(
                    false, a[mi], false, bv[ni], (short)0, acc[mi][ni],
                    false, false);
        Wp += wStep;
    }

    // Store: C/D layout — VGPR r holds rows (tile + r) and (tile + r + 8)
    #pragma unroll
    for (int mi = 0; mi < MT; ++mi) {
        #pragma unroll
        for (int ni = 0; ni < NT; ++ni) {
            const int col = nBase + 16 * ni + l16;
            const float bn = (epi == 2) ? bias[col] : 0.f;
            #pragma unroll
            for (int r = 0; r < 8; ++r) {
                const int orow = mBase + 16 * mi + r + 8 * half;
                const size_t idx = (size_t)orow * ldc + col;
                float v = acc[mi][ni][r];
                if (epi == 1) {
                    v += C[idx];
                } else if (epi == 2) {
                    float s = v + bn;
                    v = (s > 20.f) ? s : __logf(1.f + __expf(s));
                }
                C[idx] = v;
            }
        }
    }
}

// ---------------------------------------------------------------------------
// Depthwise causal conv (width 4) + SiLU.  xi = xz[..., :D_INNER]
// ---------------------------------------------------------------------------
__global__ __launch_bounds__(256) void conv_silu_kernel(
    const float* __restrict__ xz, const float* __restrict__ cw,
    const float* __restrict__ cb, float* __restrict__ xc)
{
    const int idx = blockIdx.x * 256 + threadIdx.x;     // b*L*D_INNER range
    const int d  = idx & (D_INNER - 1);
    const int bl = idx >> 11;                            // b*L + l
    const int l  = bl & (SEQ - 1);
    const int b  = bl >> 11;
    float acc = cb[d];
    #pragma unroll
    for (int j = 0; j < D_CONV; ++j) {
        const int ls = l - (D_CONV - 1) + j;
        if (ls >= 0)
            acc += cw[d * D_CONV + j] *
                   xz[((size_t)(b * SEQ + ls)) * (2 * D_INNER) + d];
    }
    xc[idx] = acc / (1.f + __expf(-acc));               // silu
}

// ---------------------------------------------------------------------------
// Selective scan + gating.  One thread = one (b,d) channel, 16 states in regs.
// B_t / C_t (shared across d) staged through LDS in chunks of 8 timesteps.
// Writes yg = (scan_y + xc*Dp) * silu(z)
// ---------------------------------------------------------------------------
#define SCAN_CHUNK 8
__global__ __launch_bounds__(256) void scan_gate_kernel(
    const float* __restrict__ dt, const float* __restrict__ xdb,
    const float* __restrict__ xc, const float* __restrict__ xz,
    const float* __restrict__ A_log, const float* __restrict__ Dp,
    float* __restrict__ yg)
{
    const int b  = blockIdx.x >> 3;                      // D_INNER/256 == 8
    const int d  = (blockIdx.x & 7) * 256 + threadIdx.x;
    float Arow[D_STATE];
    #pragma unroll
    for (int s = 0; s < D_STATE; ++s) Arow[s] = -__expf(A_log[d * D_STATE + s]);
    const float Dd = Dp[d];
    float h[D_STATE];
    #pragma unroll
    for (int s = 0; s < D_STATE; ++s) h[s] = 0.f;

    __shared__ float sBC[SCAN_CHUNK][32];                // [li][B0..15 C0..15]

    for (int l0 = 0; l0 < SEQ; l0 += SCAN_CHUNK) {
        __syncthreads();
        {
            const int t  = threadIdx.x;                  // 256 = 8*32
            const int li = t >> 5;
            const int si = t & 31;
            sBC[li][si] =
                xdb[((size_t)(b * SEQ + l0 + li)) * 96 + DT_RANK + si];
        }
        __syncthreads();
        #pragma unroll
        for (int li = 0; li < SCAN_CHUNK; ++li) {
            const int l = l0 + li;
            const size_t base = ((size_t)(b * SEQ + l)) * D_INNER + d;
            const float dtv = dt[base];
            const float xv  = xc[base];
            const float zv  =
                xz[((size_t)(b * SEQ + l)) * (2 * D_INNER) + D_INNER + d];
            float y = 0.f;
            #pragma unroll
            for (int s = 0; s < D_STATE; ++s) {
                const float dA = __expf(dtv * Arow[s]);
                h[s] = h[s] * dA + dtv * sBC[li][s] * xv;
                y += h[s] * sBC[li][D_STATE + s];
            }
            const float sg = zv / (1.f + __expf(-zv));
            yg[base] = (y + xv * Dd) * sg;
        }
    }
}

// ---------------------------------------------------------------------------
// out = x + proj_b   (GEMMs then accumulate into it)
// ---------------------------------------------------------------------------
__global__ __launch_bounds__(256) void init_out_kernel(
    const float* __restrict__ x, const float* __restrict__ pb,
    float* __restrict__ out, int total)
{
    const int i = blockIdx.x * 256 + threadIdx.x;
    if (i < total) out[i] = x[i] + pb[i & (D_MODEL - 1)];
}

// ---------------------------------------------------------------------------
extern "C" void kernel_launch(void* const* d_in, const int* in_sizes, int n_in,
                              void* d_out, int out_size, void* d_ws, size_t ws_size,
                              hipStream_t stream) {
    // setup_inputs() dict order:
    const float* x        = (const float*)d_in[0];
    const float* ln_g     = (const float*)d_in[1];
    const float* ln_b     = (const float*)d_in[2];
    const float* out_w[2] = {(const float*)d_in[3],  (const float*)d_in[4]};
    const float* proj_w   = (const float*)d_in[5];
    const float* proj_b   = (const float*)d_in[6];
    const float* in_w[2]   = {(const float*)d_in[7],  (const float*)d_in[15]};
    const float* conv_w[2] = {(const float*)d_in[8],  (const float*)d_in[16]};
    const float* conv_b[2] = {(const float*)d_in[9],  (const float*)d_in[17]};
    const float* xproj_w[2]= {(const float*)d_in[10], (const float*)d_in[18]};
    const float* dt_w[2]   = {(const float*)d_in[11], (const float*)d_in[19]};
    const float* dt_b[2]   = {(const float*)d_in[12], (const float*)d_in[20]};
    const float* A_log[2]  = {(const float*)d_in[13], (const float*)d_in[21]};
    const float* Dp[2]     = {(const float*)d_in[14], (const float*)d_in[22]};
    float* out = (float*)d_out;

    // Workspace layout (floats). Buffers for xz/xc/xdb/dt are reused between
    // the forward and backward direction (pipeline is sequential per dir).
    float* ws  = (float*)d_ws;
    float* xn  = ws;                                       //  4,194,304
    float* xz  = xn  + (size_t)BATCH * SEQ * D_MODEL;      // 16,777,216
    float* xc  = xz  + (size_t)BATCH * SEQ * 2 * D_INNER;  //  8,388,608
    float* xdb = xc  + (size_t)BATCH * SEQ * D_INNER;      //    393,216
    float* dtb = xdb + (size_t)BATCH * SEQ * 96;           //  8,388,608
    float* ygf = dtb + (size_t)BATCH * SEQ * D_INNER;      //  8,388,608
    float* ygb = ygf + (size_t)BATCH * SEQ * D_INNER;      //  8,388,608
    float* wcf = ygb + (size_t)BATCH * SEQ * D_INNER;      //  2,097,152
    float* wcb = wcf + (size_t)D_INNER * D_MODEL;          //  2,097,152
    float* yg[2] = {ygf, ygb};
    float* Wc[2] = {wcf, wcb};

    const int M = BATCH * SEQ;   // 4096 rows

    // MT=2 always (all M are multiples of 32); NT=4 when 64 | N, else NT=3.
    auto gemm = [&](const float* A, int lda, const float* W, int ldb,
                    float* C, int ldc, const float* bias,
                    int m, int n, int k, int revL, int epi) {
        if ((n & 63) == 0) {
            dim3 grid(n / 64, m / 32);
            gemm_wmma_f32<2, 4><<<grid, 32, 0, stream>>>(
                A, lda, W, ldb, C, ldc, bias, k, revL, epi);
        } else {            // n % 48 == 0 (x_proj: n = 96)
            dim3 grid(n / 48, m / 32);
            gemm_wmma_f32<2, 3><<<grid, 32, 0, stream>>>(
                A, lda, W, ldb, C, ldc, bias, k, revL, epi);
        }
    };

    // 1) LayerNorm
    layernorm_kernel<<<M, 256, 0, stream>>>(x, ln_g, ln_b, xn);

    // 2) Fold out_proj into final projection:  Wc[dir] = out_w @ proj_w_half
    gemm(out_w[0], D_MODEL, proj_w,                             D_MODEL,
         Wc[0], D_MODEL, nullptr, D_INNER, D_MODEL, D_MODEL, 0, 0);
    gemm(out_w[1], D_MODEL, proj_w + (size_t)D_MODEL * D_MODEL, D_MODEL,
         Wc[1], D_MODEL, nullptr, D_INNER, D_MODEL, D_MODEL, 0, 0);

    // 3) out = x + proj_b
    init_out_kernel<<<(M * D_MODEL) / 256, 256, 0, stream>>>(
        x, proj_b, out, M * D_MODEL);

    // 4) Both Mamba directions (backward reads xn / writes out row-reversed)
    for (int dir = 0; dir < 2; ++dir) {
        const int rev = dir ? SEQ : 0;
        // in_proj: xz = (rev?)xn @ in_w     [4096 x 1024] @ [1024 x 4096]
        gemm(xn, D_MODEL, in_w[dir], 2 * D_INNER, xz, 2 * D_INNER,
             nullptr, M, 2 * D_INNER, D_MODEL, rev, 0);
        // depthwise conv + silu -> xc
        conv_silu_kernel<<<(M * D_INNER) / 256, 256, 0, stream>>>(
            xz, conv_w[dir], conv_b[dir], xc);
        // x_proj: xdb = xc @ xproj_w        [4096 x 2048] @ [2048 x 96]
        gemm(xc, D_INNER, xproj_w[dir], 96, xdb, 96,
             nullptr, M, 96, D_INNER, 0, 0);
        // dt = softplus(xdb[:, :64] @ dt_w + dt_b)
        gemm(xdb, 96, dt_w[dir], D_INNER, dtb, D_INNER,
             dt_b[dir], M, D_INNER, DT_RANK, 0, 2);
        // selective scan + gating -> yg[dir]
        scan_gate_kernel<<<BATCH * (D_INNER / 256), 256, 0, stream>>>(
            dtb, xdb, xc, xz, A_log[dir], Dp[dir], yg[dir]);
        // out += (rev?)yg @ Wc[dir]         [4096 x 2048] @ [2048 x 1024]
        gemm(yg[dir], D_INNER, Wc[dir], D_MODEL, out, D_MODEL,
             nullptr, M, D_MODEL, D_INNER, rev, 1);
    }
}